// GNNModel_49589692399895
// MI455X (gfx1250) — compile-verified
//
#include <hip/hip_runtime.h>

typedef __attribute__((ext_vector_type(2))) float v2f;
typedef __attribute__((ext_vector_type(8))) float v8f;

// ---------------- small elementwise kernels ----------------

__global__ void set_val_kernel(float* __restrict__ p, float v, int n) {
    int i = blockIdx.x * blockDim.x + threadIdx.x;
    if (i < n) p[i] = v;
}

__global__ void deg_count_kernel(const int* __restrict__ dst, float* __restrict__ deg, int E) {
    int i = blockIdx.x * blockDim.x + threadIdx.x;
    if (i < E) unsafeAtomicAdd(&deg[dst[i]], 1.0f);
}

__global__ void rsqrt_kernel(float* __restrict__ p, int n) {
    int i = blockIdx.x * blockDim.x + threadIdx.x;
    if (i < n) p[i] = rsqrtf(p[i]);
}

__global__ void bias_add_kernel(float* __restrict__ out, const float* __restrict__ b,
                                int n, int mask) {
    int i = blockIdx.x * blockDim.x + threadIdx.x;
    if (i < n) out[i] += b[i & mask];
}

// ---------------- WMMA f32 GEMM: C[M x FOUT] = act(A)[M x FIN] * W[FIN x FOUT] ----------------
// One wave32 computes a full 16-row x FOUT-col stripe: NT = FOUT/16 accumulators.
// A registers are loaded once per k-step and reused across all NT column tiles.
// V_WMMA_F32_16X16X4_F32 lane layout (16x4 A / 4x16 B, 2 VGPRs each):
//   lanes 0-15  : K = 0,1 (v0,v1), M/N = lane
//   lanes 16-31 : K = 2,3 (v0,v1), M/N = lane-16
// C/D (8 VGPRs): vgpr g -> row (g + 8*half), col = lane&15.
template <int FIN, int FOUT, bool FUSE_BIAS_RELU>
__global__ __launch_bounds__(256) void gemm_wmma_kernel(
    const float* __restrict__ A, const float* __restrict__ W,
    const float* __restrict__ bias, float* __restrict__ C,
    int M, int mtiles) {
    constexpr int NT = FOUT / 16;
    int wave = blockIdx.x * (blockDim.x >> 5) + (threadIdx.x >> 5);
    if (wave >= mtiles) return;  // wave-uniform: EXEC stays all-ones for WMMA

    const int row0 = wave * 16;
    const int lane = threadIdx.x & 31;
    const int half = lane >> 4;   // 0: K base 0, 1: K base 2
    const int lm   = lane & 15;
    const int koff = half * 2;

    const float* __restrict__ arow = A + (size_t)(row0 + lm) * FIN;
    const bool full_tile = (row0 + 16) <= M;
    // For a partial last tile, clamp the A row pointer to a valid row; the
    // garbage rows only influence output rows that are never stored.
    const float* __restrict__ arow_safe =
        full_tile ? arow : (A + (size_t)((row0 + lm) < M ? (row0 + lm) : (M - 1)) * FIN);

    v8f acc[NT];
#pragma unroll
    for (int t = 0; t < NT; ++t) acc[t] = v8f{};

#pragma unroll 4
    for (int k = 0; k < FIN; k += 4) {
        float a0 = arow_safe[k + koff];
        float a1 = arow_safe[k + koff + 1];
        if (FUSE_BIAS_RELU) {
            a0 = fmaxf(a0 + bias[k + koff], 0.0f);
            a1 = fmaxf(a1 + bias[k + koff + 1], 0.0f);
        }
        v2f a;
        a.x = a0;
        a.y = a1;
#pragma unroll
        for (int t = 0; t < NT; ++t) {
            v2f b;
            b.x = W[(size_t)(k + koff) * FOUT + t * 16 + lm];
            b.y = W[(size_t)(k + koff + 1) * FOUT + t * 16 + lm];
            acc[t] = __builtin_amdgcn_wmma_f32_16x16x4_f32(
                /*neg_a=*/false, a, /*neg_b=*/false, b,
                /*c_mod=*/(short)0, acc[t], /*reuse_a=*/false, /*reuse_b=*/false);
        }
    }

    if (full_tile) {
        // straight-line store epilogue, no exec juggling
#pragma unroll
        for (int g = 0; g < 8; ++g) {
            float* crow = C + (size_t)(row0 + half * 8 + g) * FOUT + lm;
#pragma unroll
            for (int t = 0; t < NT; ++t) crow[t * 16] = acc[t][g];
        }
    } else {
#pragma unroll
        for (int g = 0; g < 8; ++g) {
            int r = row0 + half * 8 + g;
            if (r < M) {
                float* crow = C + (size_t)r * FOUT + lm;
#pragma unroll
                for (int t = 0; t < NT; ++t) crow[t * 16] = acc[t][g];
            }
        }
    }
}

// ---------------- edge aggregation: out[d] += dinv[s]*dinv[d] * h[s] ----------------
// Edges e in [0,E) from src/dst arrays; e in [E,E+N) are self loops (s=d=e-E).
// Each thread handles 4 consecutive features (float4 gather + 4 f32 global atomics).
template <int F>
__global__ __launch_bounds__(256) void aggregate_kernel(
    const float* __restrict__ h, const int* __restrict__ src,
    const int* __restrict__ dst, const float* __restrict__ dinv,
    float* __restrict__ out, int E, int N) {
    const int TPE = F / 4;
    long long idx = (long long)blockIdx.x * blockDim.x + threadIdx.x;
    long long total = (long long)(E + N) * TPE;
    if (idx >= total) return;

    int e  = (int)(idx / TPE);
    int f4 = (int)(idx % TPE) * 4;

    int s, d;
    if (e < E) {
        s = src[e];
        d = dst[e];
    } else {
        s = d = e - E;
    }
    float norm = dinv[s] * dinv[d];

    const float4 hv = *reinterpret_cast<const float4*>(h + (size_t)s * F + f4);
    float* o = out + (size_t)d * F + f4;
    unsafeAtomicAdd(o + 0, hv.x * norm);
    unsafeAtomicAdd(o + 1, hv.y * norm);
    unsafeAtomicAdd(o + 2, hv.z * norm);
    unsafeAtomicAdd(o + 3, hv.w * norm);
}

// ---------------- launch ----------------

extern "C" void kernel_launch(void* const* d_in, const int* in_sizes, int n_in,
                              void* d_out, int out_size, void* d_ws, size_t ws_size,
                              hipStream_t stream) {
    const float* x  = (const float*)d_in[0];
    const int*   ei = (const int*)d_in[1];
    const float* W1 = (const float*)d_in[2];
    const float* b1 = (const float*)d_in[3];
    const float* W2 = (const float*)d_in[4];
    const float* b2 = (const float*)d_in[5];

    const int FIN = 128, F1 = 64, F2 = 32;
    const int N = in_sizes[0] / FIN;
    const int E = in_sizes[1] / 2;
    const int* src = ei;
    const int* dst = ei + E;

    float* ws   = (float*)d_ws;
    float* deg  = ws;                          // N floats (becomes dinv in place)
    float* h1   = ws + N;                      // N*F1 floats (reused as h2: N*F2)
    float* agg1 = ws + N + (size_t)N * F1;     // N*F1 floats
    float* outp = (float*)d_out;               // N*F2 (doubles as agg2 accumulator)

    hipMemsetAsync(agg1, 0, (size_t)N * F1 * sizeof(float), stream);
    hipMemsetAsync(outp, 0, (size_t)N * F2 * sizeof(float), stream);

    const int thr = 256;
    const int wpb = thr / 32;

    // degrees (self loop contributes 1.0 via init) -> dinv
    set_val_kernel<<<(N + thr - 1) / thr, thr, 0, stream>>>(deg, 1.0f, N);
    deg_count_kernel<<<(E + thr - 1) / thr, thr, 0, stream>>>(dst, deg, E);
    rsqrt_kernel<<<(N + thr - 1) / thr, thr, 0, stream>>>(deg, N);

    // layer 1 GEMM: h1 = x @ W1
    {
        int mtiles = (N + 15) / 16;
        int blocks = (mtiles + wpb - 1) / wpb;
        gemm_wmma_kernel<128, 64, false>
            <<<blocks, thr, 0, stream>>>(x, W1, nullptr, h1, N, mtiles);
    }
    // layer 1 aggregation -> agg1
    {
        long long total = (long long)(E + N) * (F1 / 4);
        int blocks = (int)((total + thr - 1) / thr);
        aggregate_kernel<64><<<blocks, thr, 0, stream>>>(h1, src, dst, deg, agg1, E, N);
    }
    // layer 2 GEMM with fused (+b1, ReLU) on A operand: h2 = relu(agg1 + b1) @ W2
    {
        int mtiles = (N + 15) / 16;
        int blocks = (mtiles + wpb - 1) / wpb;
        gemm_wmma_kernel<64, 32, true>
            <<<blocks, thr, 0, stream>>>(agg1, W2, b1, h1 /*h2*/, N, mtiles);
    }
    // layer 2 aggregation -> d_out (zeroed above)
    {
        long long total = (long long)(E + N) * (F2 / 4);
        int blocks = (int)((total + thr - 1) / thr);
        aggregate_kernel<32><<<blocks, thr, 0, stream>>>(h1, src, dst, deg, outp, E, N);
    }
    // final bias
    bias_add_kernel<<<(N * F2 + thr - 1) / thr, thr, 0, stream>>>(outp, b2, N * F2, F2 - 1);
}